// StockPredict_1949915152743
// MI455X (gfx1250) — compile-verified
//
#include <hip/hip_runtime.h>
#include <hip/hip_bf16.h>
#include <math.h>

// ---------------------------------------------------------------------------
// StockPredict on MI455X (gfx1250): fused time-mixer in LDS + WMMA fp32 GEMMs
// + async global->LDS staging + pre-transposed conv weights for coalesced B.
// N=2048 stocks, T=32, C=512, TC=56, H=8, DH=64, 64 industries.
// ---------------------------------------------------------------------------

typedef float v2f __attribute__((ext_vector_type(2)));
typedef float v8f __attribute__((ext_vector_type(8)));

#define NSTK 2048
#define CH   512
#define TT   32
#define TCAT 56
#define GMAX 256   // max industry group size we support (E[g]=32, P(g>96)~0)

static __device__ inline v8f wmma4(v2f a, v2f b, v8f c) {
  // D = A(16x4,f32) * B(4x16,f32) + C(16x16,f32)
  return __builtin_amdgcn_wmma_f32_16x16x4_f32(
      /*neg_a=*/false, a, /*neg_b=*/false, b,
      /*c_mod=*/(short)0, c, /*reuse_a=*/false, /*reuse_b=*/false);
}

// Async copy 16B global -> LDS (no VGPR roundtrip, ASYNCcnt-tracked).
static __device__ inline void async_ld_b128(unsigned lds_off, const void* gaddr) {
  asm volatile("global_load_async_to_lds_b128 %0, %1, off"
               :: "v"(lds_off), "v"(gaddr) : "memory");
}
static __device__ inline void wait_async0() {
  asm volatile("s_wait_asynccnt 0x0" ::: "memory");
}

static __device__ inline float warpsum(float v) {
#pragma unroll
  for (int o = 16; o > 0; o >>= 1) v += __shfl_xor(v, o, 32);
  return v;
}

// LayerNorm over a whole [ts, C] slab in LDS (jointly over both dims).
static __device__ inline void ln2d(float* slab, int nelem,
                                   const float* __restrict__ g,
                                   const float* __restrict__ b,
                                   float* red1, float* red2, int tid) {
  float s = 0.f, ss = 0.f;
  for (int e = tid; e < nelem; e += 256) { float v = slab[e]; s += v; ss += v * v; }
  s = warpsum(s); ss = warpsum(ss);
  int wid = tid >> 5, lane = tid & 31;
  if (lane == 0) { red1[wid] = s; red2[wid] = ss; }
  __syncthreads();
  float ts = 0.f, tss = 0.f;
#pragma unroll
  for (int i = 0; i < 8; ++i) { ts += red1[i]; tss += red2[i]; }
  float m   = ts / (float)nelem;
  float var = tss / (float)nelem - m * m;
  float inv = rsqrtf(var + 1e-5f);
  __syncthreads();
  for (int e = tid; e < nelem; e += 256)
    slab[e] = (slab[e] - m) * inv * g[e] + b[e];
}

// ---------------------------------------------------------------------------
// Kernel 0: transpose conv weights once: wT[k_flat][c_out] so WMMA B-fragment
// loads are lane-contiguous (64B bursts) instead of 4KB-strided per lane.
// grid.x: 0..1023 -> w1 rows, 1024..3071 -> w2 rows.
// ---------------------------------------------------------------------------
__global__ __launch_bounds__(256)
void k_wtrans(const float* __restrict__ w1, const float* __restrict__ w2,
              float* __restrict__ w1t, float* __restrict__ w2t) {
  int bk = blockIdx.x, tid = threadIdx.x;
  if (bk < 1024) {             // w1: [512][1024] -> w1t: [1024][512]
    int k = bk;
    for (int co = tid; co < CH; co += 256)
      w1t[(size_t)k * CH + co] = w1[(size_t)co * 1024 + k];
  } else {                     // w2: [512][2048] -> w2t: [2048][512]
    int k = bk - 1024;
    for (int co = tid; co < CH; co += 256) {
      w2t[(size_t)(2 * k + 0) * CH + co] = w2[(size_t)co * 2048 + 2 * k + 0];
      w2t[(size_t)(2 * k + 1) * CH + co] = w2[(size_t)co * 2048 + 2 * k + 1];
    }
  }
}

// ---------------------------------------------------------------------------
// Kernel 1: per-stock fused time mixer. One block (8 wave32) per stock.
//   LDS: cat[56][512] (x | f1 | f2), hbf[512][56] (TriU1 out), tril weights.
//   conv1: GEMM M=16,K=1024,N=512 ; conv2: M=16(8 valid),K=2048,N=512
//   TriU:  H = Z[512x56] * tril(W)^T[56x56]  -> 32x4 tiles, K=56 (14 chunks)
//   pooled[n,c] = mean_s h2[c,s]
// ---------------------------------------------------------------------------
__global__ __launch_bounds__(256)
void k_timemixer(const float* __restrict__ x,
                 const float* __restrict__ w1t, const float* __restrict__ cb1,
                 const float* __restrict__ w2t, const float* __restrict__ cb2,
                 const float* __restrict__ g1, const float* __restrict__ bb1,
                 const float* __restrict__ g2, const float* __restrict__ bb2,
                 const float* __restrict__ tw1, const float* __restrict__ tb1,
                 const float* __restrict__ tw2, const float* __restrict__ tb2,
                 float* __restrict__ pooled) {
  __shared__ float cat[TCAT * CH];   // 114688 B
  __shared__ float hbf[CH * TCAT];   // 114688 B
  __shared__ float wtri[64 * TCAT];  // 14336 B (rows 56..63 zero)
  __shared__ float red1[8], red2[8];

  const int tid = threadIdx.x;
  const int wid = tid >> 5, lane = tid & 31;
  const int half = lane >> 4, r = lane & 15;
  const int n = blockIdx.x;

  // --- async-stage x[n] into cat rows 0..31 (global -> LDS direct) --------
  {
    const float4* xg = (const float4*)(x + (size_t)n * TT * CH);
    float4* c4 = (float4*)cat;
    for (int i = tid; i < TT * CH / 4; i += 256)
      async_ld_b128((unsigned)(size_t)(c4 + i), xg + i);
    wait_async0();
  }
  __syncthreads();

  // --- conv1 (K=2, stride 2) -> cat rows 32..47 ---------------------------
  for (int nt = 0; nt < 4; ++nt) {
    int cb = (wid * 4 + nt) * 16;
    int col = cb + r;
    v8f acc = {0, 0, 0, 0, 0, 0, 0, 0};
    for (int q = 0; q < 256; ++q) {           // K chunk: k_idx 4q..4q+3
      int ci = 2 * q + half;                  // c_in for this lane half
      v2f a, b;
      a.x = cat[(2 * r + 0) * CH + ci];       // x[2t+0, ci]
      a.y = cat[(2 * r + 1) * CH + ci];       // x[2t+1, ci]
      const float* wr = w1t + (size_t)(4 * q + 2 * half) * CH + col;
      b.x = wr[0]; b.y = wr[CH];              // rows k, k+1 (lane-coalesced)
      if ((q & 15) == 0 && 4 * q + 64 < 1024)
        __builtin_prefetch(w1t + (size_t)(4 * q + 64) * CH + col, 0, 1);
      acc = wmma4(a, b, acc);
    }
    float bias = cb1[col];
#pragma unroll
    for (int vg = 0; vg < 8; ++vg)
      cat[(32 + vg + 8 * half) * CH + col] = acc[vg] + bias;
  }
  __syncthreads();

  // --- conv2 (K=4, stride 4) -> cat rows 48..55 ---------------------------
  for (int nt = 0; nt < 4; ++nt) {
    int cb = (wid * 4 + nt) * 16;
    int col = cb + r;
    v8f acc = {0, 0, 0, 0, 0, 0, 0, 0};
    for (int q = 0; q < 512; ++q) {           // c_in = q, kk = 2*half..+1
      int kk = 2 * half;
      v2f a, b;
      a.x = (r < 8) ? cat[(4 * r + kk + 0) * CH + q] : 0.f;
      a.y = (r < 8) ? cat[(4 * r + kk + 1) * CH + q] : 0.f;
      const float* wr = w2t + (size_t)(4 * q + kk) * CH + col;
      b.x = wr[0]; b.y = wr[CH];
      if ((q & 15) == 0 && 4 * q + 64 < 2048)
        __builtin_prefetch(w2t + (size_t)(4 * q + 64) * CH + col, 0, 1);
      acc = wmma4(a, b, acc);
    }
    if (half == 0) {                          // only rows M=0..7 valid
      float bias = cb2[col];
#pragma unroll
      for (int vg = 0; vg < 8; ++vg)
        cat[(48 + vg) * CH + col] = acc[vg] + bias;
    }
  }
  __syncthreads();

  // --- LayerNorms over [16,512] and [8,512] slabs -------------------------
  ln2d(cat + 32 * CH, 16 * CH, g1, bb1, red1, red2, tid);
  __syncthreads();
  ln2d(cat + 48 * CH, 8 * CH, g2, bb2, red1, red2, tid);
  __syncthreads();

  // --- TriU1: H = Z * tril(W1)^T, +b1, hardswish -> hbf[c][s] -------------
  for (int i = tid; i < 64 * TCAT; i += 256) {
    int s = i / TCAT, t = i - s * TCAT;
    wtri[i] = (s < TCAT && s >= t) ? tw1[s * TCAT + t] : 0.f;
  }
  __syncthreads();
  for (int tt = 0; tt < 16; ++tt) {
    int tile = wid * 16 + tt;
    int cm = (tile >> 2) * 16, sb = (tile & 3) * 16;
    v8f acc = {0, 0, 0, 0, 0, 0, 0, 0};
    for (int q = 0; q < 14; ++q) {
      int t0 = 4 * q + 2 * half;
      v2f a, b;
      a.x = cat[(t0 + 0) * CH + cm + r];      // Z[c, t] = cat[t][c]
      a.y = cat[(t0 + 1) * CH + cm + r];
      b.x = wtri[(sb + r) * TCAT + t0];
      b.y = wtri[(sb + r) * TCAT + t0 + 1];
      acc = wmma4(a, b, acc);
    }
    int s = sb + r;
    if (s < TCAT) {
      float bias = tb1[s];
#pragma unroll
      for (int vg = 0; vg < 8; ++vg) {
        float v = acc[vg] + bias;
        v = v * fminf(fmaxf(v + 3.f, 0.f), 6.f) * (1.f / 6.f);  // hardswish
        hbf[(cm + vg + 8 * half) * TCAT + s] = v;
      }
    }
  }
  __syncthreads();

  // --- TriU2: H2 = H * tril(W2)^T + b2 -> cat (reused flat [512][56]) -----
  for (int i = tid; i < 64 * TCAT; i += 256) {
    int s = i / TCAT, t = i - s * TCAT;
    wtri[i] = (s < TCAT && s >= t) ? tw2[s * TCAT + t] : 0.f;
  }
  __syncthreads();
  for (int tt = 0; tt < 16; ++tt) {
    int tile = wid * 16 + tt;
    int cm = (tile >> 2) * 16, sb = (tile & 3) * 16;
    v8f acc = {0, 0, 0, 0, 0, 0, 0, 0};
    for (int q = 0; q < 14; ++q) {
      int t0 = 4 * q + 2 * half;
      v2f a, b;
      a.x = hbf[(cm + r) * TCAT + t0];
      a.y = hbf[(cm + r) * TCAT + t0 + 1];
      b.x = wtri[(sb + r) * TCAT + t0];
      b.y = wtri[(sb + r) * TCAT + t0 + 1];
      acc = wmma4(a, b, acc);
    }
    int s = sb + r;
    if (s < TCAT) {
      float bias = tb2[s];
#pragma unroll
      for (int vg = 0; vg < 8; ++vg)
        cat[(cm + vg + 8 * half) * TCAT + s] = acc[vg] + bias;
    }
  }
  __syncthreads();

  // --- mean pool over s ---------------------------------------------------
  for (int c = tid; c < CH; c += 256) {
    float sm = 0.f;
#pragma unroll 8
    for (int s = 0; s < TCAT; ++s) sm += cat[c * TCAT + s];
    pooled[(size_t)n * CH + c] = sm * (1.f / (float)TCAT);
  }
}

// ---------------------------------------------------------------------------
// Kernel 2: Q/K/V projections.  [2048x512]x[512x512] via WMMA fp32.
// One block per 16-row tile; 8 waves cover 3 matrices x 32 col tiles.
// (wq/wk/wv are [k][j] so B loads are already lane-coalesced.)
// ---------------------------------------------------------------------------
__global__ __launch_bounds__(256)
void k_qkv(const float* __restrict__ pooled,
           const float* __restrict__ wq, const float* __restrict__ bq,
           const float* __restrict__ wk, const float* __restrict__ bk,
           const float* __restrict__ wv, const float* __restrict__ bv,
           float* __restrict__ qo, float* __restrict__ ko, float* __restrict__ vo) {
  __shared__ float als[16 * CH];  // A tile staged once, reused by 96 tiles
  const int tid = threadIdx.x;
  const int wid = tid >> 5, lane = tid & 31;
  const int half = lane >> 4, r = lane & 15;
  const int m0 = blockIdx.x * 16;

  {
    const float4* ag = (const float4*)(pooled + (size_t)m0 * CH);
    float4* s4 = (float4*)als;
    for (int i = tid; i < 16 * CH / 4; i += 256)
      async_ld_b128((unsigned)(size_t)(s4 + i), ag + i);
    wait_async0();
  }
  __syncthreads();

  for (int t = 0; t < 12; ++t) {
    int g = wid * 12 + t;
    int mat = g >> 5, nt = g & 31, jb = nt * 16;
    const float* w  = (mat == 0) ? wq : (mat == 1) ? wk : wv;
    const float* bs = (mat == 0) ? bq : (mat == 1) ? bk : bv;
    float*       o  = (mat == 0) ? qo : (mat == 1) ? ko : vo;
    v8f acc = {0, 0, 0, 0, 0, 0, 0, 0};
    for (int q = 0; q < 128; ++q) {
      int kr = 4 * q + 2 * half;
      v2f a, b;
      a.x = als[r * CH + kr];
      a.y = als[r * CH + kr + 1];
      b.x = w[(size_t)(kr + 0) * CH + jb + r];
      b.y = w[(size_t)(kr + 1) * CH + jb + r];
      if ((q & 15) == 0 && kr + 64 < CH)
        __builtin_prefetch(&w[(size_t)(kr + 64) * CH + jb + r], 0, 1);
      acc = wmma4(a, b, acc);
    }
    int col = jb + r; float bias = bs[col];
#pragma unroll
    for (int vg = 0; vg < 8; ++vg) {
      int row = m0 + vg + 8 * half;
      o[(size_t)row * CH + col] = acc[vg] + bias;
    }
  }
}

// ---------------------------------------------------------------------------
// Kernel 3: deterministic industry bucketing (counts, offsets, members).
// rank(n) = #{m < n : ind[m]==ind[n]}  -> stable order, reproducible fp sums.
// ---------------------------------------------------------------------------
__global__ __launch_bounds__(256)
void k_groups(const int* __restrict__ ind, int* __restrict__ counts,
              int* __restrict__ offs, int* __restrict__ members) {
  __shared__ int scnt[64];
  __shared__ int soff[64];
  int tid = threadIdx.x;
  if (tid < 64) scnt[tid] = 0;
  __syncthreads();
  for (int n = tid; n < NSTK; n += 256) atomicAdd(&scnt[ind[n]], 1);
  __syncthreads();
  if (tid == 0) {
    int a = 0;
    for (int i = 0; i < 64; ++i) { soff[i] = a; a += scnt[i]; }
  }
  __syncthreads();
  if (tid < 64) { counts[tid] = scnt[tid]; offs[tid] = soff[tid]; }
  for (int n = tid; n < NSTK; n += 256) {
    int id = ind[n], rank = 0;
    for (int m = 0; m < n; ++m) rank += (ind[m] == id) ? 1 : 0;
    members[soff[id] + rank] = n;
  }
}

// ---------------------------------------------------------------------------
// Kernel 4: block-sparse attention within each industry (flash-style online
// softmax). One block per industry; per head stage K,V for the group in LDS.
// Output overwrites the Q buffer in-place (each row's q consumed first).
// ---------------------------------------------------------------------------
__global__ __launch_bounds__(256)
void k_attn(float* __restrict__ qbuf, const float* __restrict__ kbuf,
            const float* __restrict__ vbuf, const int* __restrict__ members,
            const int* __restrict__ counts, const int* __restrict__ offs) {
  __shared__ float kh[GMAX * 65];  // pad 65 to avoid bank conflicts
  __shared__ float vh[GMAX * 65];
  const int tid = threadIdx.x, wid = tid >> 5, lane = tid & 31;
  const int indus = blockIdx.x;
  int g = counts[indus]; if (g > GMAX) g = GMAX;
  const int base = offs[indus];
  if (g == 0) return;

  for (int h = 0; h < 8; ++h) {
    for (int i = tid; i < g * 64; i += 256) {
      int j = i >> 6, d = i & 63;
      int m = members[base + j];
      kh[j * 65 + d] = kbuf[(size_t)m * CH + h * 64 + d];
      vh[j * 65 + d] = vbuf[(size_t)m * CH + h * 64 + d];
    }
    __syncthreads();
    for (int jq = wid; jq < g; jq += 8) {
      int nq = members[base + jq];
      float qa = qbuf[(size_t)nq * CH + h * 64 + lane];
      float qb = qbuf[(size_t)nq * CH + h * 64 + lane + 32];
      float mrun = -3.4e38f, lsum = 0.f, oa = 0.f, ob = 0.f;
      for (int j = 0; j < g; ++j) {
        float part = qa * kh[j * 65 + lane] + qb * kh[j * 65 + lane + 32];
        float s = warpsum(part) * 0.125f;        // 1/sqrt(64)
        float nm = fmaxf(mrun, s);
        float sc = __expf(mrun - nm);
        float p  = __expf(s - nm);
        lsum = lsum * sc + p;
        oa = oa * sc + p * vh[j * 65 + lane];
        ob = ob * sc + p * vh[j * 65 + lane + 32];
        mrun = nm;
      }
      float rl = 1.f / lsum;
      qbuf[(size_t)nq * CH + h * 64 + lane]      = oa * rl;
      qbuf[(size_t)nq * CH + h * 64 + lane + 32] = ob * rl;
    }
    __syncthreads();
  }
}

// ---------------------------------------------------------------------------
// Kernel 5: output projection + residual: h = pooled + attn @ wo + bo.
// ---------------------------------------------------------------------------
__global__ __launch_bounds__(256)
void k_oproj(const float* __restrict__ attn, const float* __restrict__ wo,
             const float* __restrict__ bo, const float* __restrict__ pooled,
             float* __restrict__ hout) {
  __shared__ float als[16 * CH];
  const int tid = threadIdx.x;
  const int wid = tid >> 5, lane = tid & 31;
  const int half = lane >> 4, r = lane & 15;
  const int m0 = blockIdx.x * 16;

  {
    const float4* ag = (const float4*)(attn + (size_t)m0 * CH);
    float4* s4 = (float4*)als;
    for (int i = tid; i < 16 * CH / 4; i += 256)
      async_ld_b128((unsigned)(size_t)(s4 + i), ag + i);
    wait_async0();
  }
  __syncthreads();

  for (int t = 0; t < 4; ++t) {
    int nt = wid * 4 + t, jb = nt * 16;
    v8f acc = {0, 0, 0, 0, 0, 0, 0, 0};
    for (int q = 0; q < 128; ++q) {
      int kr = 4 * q + 2 * half;
      v2f a, b;
      a.x = als[r * CH + kr];
      a.y = als[r * CH + kr + 1];
      b.x = wo[(size_t)(kr + 0) * CH + jb + r];
      b.y = wo[(size_t)(kr + 1) * CH + jb + r];
      if ((q & 15) == 0 && kr + 64 < CH)
        __builtin_prefetch(&wo[(size_t)(kr + 64) * CH + jb + r], 0, 1);
      acc = wmma4(a, b, acc);
    }
    int col = jb + r; float bias = bo[col];
#pragma unroll
    for (int vg = 0; vg < 8; ++vg) {
      int row = m0 + vg + 8 * half;
      hout[(size_t)row * CH + col] = acc[vg] + bias + pooled[(size_t)row * CH + col];
    }
  }
}

// ---------------------------------------------------------------------------
// Kernel 6: final projection out[n] = h[n,:] . wp + bp  (one wave per row)
// ---------------------------------------------------------------------------
__global__ __launch_bounds__(256)
void k_final(const float* __restrict__ h, const float* __restrict__ wp,
             const float* __restrict__ bp, float* __restrict__ out) {
  const int tid = threadIdx.x, wid = tid >> 5, lane = tid & 31;
  const int row = blockIdx.x * 8 + wid;
  float part = 0.f;
  for (int c = lane; c < CH; c += 32) part += h[(size_t)row * CH + c] * wp[c];
  float s = warpsum(part);
  if (lane == 0) out[row] = s + bp[0];
}

// ---------------------------------------------------------------------------
extern "C" void kernel_launch(void* const* d_in, const int* in_sizes, int n_in,
                              void* d_out, int out_size, void* d_ws, size_t ws_size,
                              hipStream_t stream) {
  (void)in_sizes; (void)n_in; (void)out_size; (void)ws_size;
  const float* x   = (const float*)d_in[0];
  const int*   ind = (const int*)d_in[1];
  const float* w1  = (const float*)d_in[2];
  const float* cb1 = (const float*)d_in[3];
  const float* w2  = (const float*)d_in[4];
  const float* cb2 = (const float*)d_in[5];
  const float* g1  = (const float*)d_in[6];
  const float* b1  = (const float*)d_in[7];
  const float* g2  = (const float*)d_in[8];
  const float* b2  = (const float*)d_in[9];
  const float* tw1 = (const float*)d_in[10];
  const float* tb1 = (const float*)d_in[11];
  const float* tw2 = (const float*)d_in[12];
  const float* tb2 = (const float*)d_in[13];
  const float* wq  = (const float*)d_in[14];
  const float* bq  = (const float*)d_in[15];
  const float* wk  = (const float*)d_in[16];
  const float* bk  = (const float*)d_in[17];
  const float* wv  = (const float*)d_in[18];
  const float* bv  = (const float*)d_in[19];
  const float* wo  = (const float*)d_in[20];
  const float* bo  = (const float*)d_in[21];
  const float* wp  = (const float*)d_in[22];
  const float* bp  = (const float*)d_in[23];
  float* out = (float*)d_out;

  // Workspace layout (4 MiB fp32 buffers; qbuf doubles as attention output,
  // kbuf doubles as the post-residual h buffer; transposed conv weights last).
  char* ws = (char*)d_ws;
  float* pooled = (float*)(ws + (size_t)0);
  float* qbuf   = (float*)(ws + (size_t)4  * 1024 * 1024);
  float* kbuf   = (float*)(ws + (size_t)8  * 1024 * 1024);
  float* vbuf   = (float*)(ws + (size_t)12 * 1024 * 1024);
  int*   counts = (int*)  (ws + (size_t)16 * 1024 * 1024);
  int*   offs    = counts + 64;
  int*   members = counts + 128;
  float* w1t    = (float*)(ws + (size_t)17 * 1024 * 1024);  // [1024][512] = 2 MiB
  float* w2t    = (float*)(ws + (size_t)19 * 1024 * 1024);  // [2048][512] = 4 MiB

  k_wtrans<<<dim3(3072), dim3(256), 0, stream>>>(w1, w2, w1t, w2t);
  k_timemixer<<<dim3(NSTK), dim3(256), 0, stream>>>(
      x, w1t, cb1, w2t, cb2, g1, b1, g2, b2, tw1, tb1, tw2, tb2, pooled);
  k_qkv<<<dim3(NSTK / 16), dim3(256), 0, stream>>>(
      pooled, wq, bq, wk, bk, wv, bv, qbuf, kbuf, vbuf);
  k_groups<<<dim3(1), dim3(256), 0, stream>>>(ind, counts, offs, members);
  k_attn<<<dim3(64), dim3(256), 0, stream>>>(qbuf, kbuf, vbuf, members, counts, offs);
  k_oproj<<<dim3(NSTK / 16), dim3(256), 0, stream>>>(qbuf, wo, bo, pooled, kbuf);
  k_final<<<dim3(NSTK / 8), dim3(256), 0, stream>>>(kbuf, wp, bp, out);
}